// SimplfiedLayerNorm_66949950210077
// MI455X (gfx1250) — compile-verified
//
#include <hip/hip_runtime.h>
#include <hip/hip_bf16.h>

// out[row][c] = x[row][c] - sum_c(x[row][:])   for 65536 rows of 512 f32.
//
// Memory-bound kernel (0.25 flop/byte): ~256 MB HBM traffic -> ~11us floor at
// 23.3 TB/s. Row sums are computed with V_WMMA_F32_16X16X4_F32 using an
// all-ones A matrix: D = ones(16x4) x B(4x16) + C gives per-column sums of B.
// Mapping B's N dimension to 16 data rows makes D's VGPR0 hold, in lane L,
// the running sum of row (L%16) -- the broadcast lands exactly where the
// output pass needs it, with zero cross-lane shuffles.

typedef float v2f __attribute__((ext_vector_type(2)));
typedef float v4f __attribute__((ext_vector_type(4)));   // native vector: OK for nontemporal builtins
typedef float v8f __attribute__((ext_vector_type(8)));

#define D_DIM 512  // feature dim (row length in floats)

__global__ __launch_bounds__(256)
void SimplfiedLayerNorm_66949950210077_kernel(const float* __restrict__ x,
                                              float* __restrict__ out,
                                              int nrows) {
    const int lane   = threadIdx.x & 31;   // wave32
    const int waveId = threadIdx.x >> 5;   // 8 waves / block
    const int stripe = blockIdx.x * 8 + waveId;
    const int r0     = stripe * 16;
    if (r0 >= nrows) return;               // wave-uniform guard: EXEC stays all-1s

    // Lane L and lane L+16 cooperate on row r0 + (L%16): lo half covers the
    // first 4 columns of each 8-column chunk, hi half the last 4 (b128 loads,
    // lane pair covers 32 contiguous bytes of one row).
    const int row = r0 + (lane & 15);
    const int hi  = lane >> 4;             // 0 for lanes 0-15, 1 for 16-31

    const v4f* __restrict__ xrow = (const v4f*)(x  + (size_t)row * D_DIM);
    v4f*       __restrict__ orow = (v4f*)      (out + (size_t)row * D_DIM);

    // A = all-ones 16x4 f32 matrix: layout-independent (every element 1.0).
    v2f ones; ones[0] = 1.0f; ones[1] = 1.0f;

    // Two accumulators to break the WMMA->WMMA RAW chain on C.
    v8f acc0 = {0.f,0.f,0.f,0.f,0.f,0.f,0.f,0.f};
    v8f acc1 = {0.f,0.f,0.f,0.f,0.f,0.f,0.f,0.f};

    // Pass 1: 64 x (b128 load + pairwise fold + WMMA accumulate).
    // Fold pairs ADJACENT register pairs ({v0,v1}+{v2,v3}) so it lowers to a
    // single v_pk_add_f32 with no register shuffling; the K-slot assignment
    // is arbitrary since A is all-ones and everything is summed over K.
#pragma unroll 8
    for (int c8 = 0; c8 < 64; c8 += 2) {
        v4f v0 = xrow[(c8 + 0) * 2 + hi];
        v4f v1 = xrow[(c8 + 1) * 2 + hi];
        v2f b0; b0[0] = v0[0] + v0[2]; b0[1] = v0[1] + v0[3];
        v2f b1; b1[0] = v1[0] + v1[2]; b1[1] = v1[1] + v1[3];
        acc0 = __builtin_amdgcn_wmma_f32_16x16x4_f32(
            /*neg_a=*/false, ones, /*neg_b=*/false, b0,
            /*c_mod=*/(short)0, acc0, /*reuse_a=*/false, /*reuse_b=*/false);
        acc1 = __builtin_amdgcn_wmma_f32_16x16x4_f32(
            /*neg_a=*/false, ones, /*neg_b=*/false, b1,
            /*c_mod=*/(short)0, acc1, /*reuse_a=*/false, /*reuse_b=*/false);
    }

    // D VGPR0, lane L = colsum(N = L%16) = full row sum of row r0 + (L%16):
    // exactly this lane's row. No shuffle needed.
    const float rsum = acc0[0] + acc1[0];

    // Pass 2: re-read (hot in WGP$/L2) and write x - rowsum. Non-temporal
    // stores keep the 128 MB output stream from evicting the x lines.
#pragma unroll 8
    for (int c8 = 0; c8 < 64; ++c8) {
        v4f v = __builtin_nontemporal_load(&xrow[c8 * 2 + hi]);
        v4f o;
        o[0] = v[0] - rsum;
        o[1] = v[1] - rsum;
        o[2] = v[2] - rsum;
        o[3] = v[3] - rsum;
        __builtin_nontemporal_store(o, &orow[c8 * 2 + hi]);
    }
}

extern "C" void kernel_launch(void* const* d_in, const int* in_sizes, int n_in,
                              void* d_out, int out_size, void* d_ws, size_t ws_size,
                              hipStream_t stream) {
    const float* x = (const float*)d_in[0];
    float* out = (float*)d_out;

    const int n       = in_sizes[0];        // 8*8192*512 = 33,554,432
    const int nrows   = n / D_DIM;          // 65,536
    const int stripes = (nrows + 15) / 16;  // 4,096 waves
    const int blocks  = (stripes + 7) / 8;  // 512 blocks x 256 threads (8 waves)

    SimplfiedLayerNorm_66949950210077_kernel<<<blocks, 256, 0, stream>>>(x, out, nrows);
}